// LogisticModel_13305808683644
// MI455X (gfx1250) — compile-verified
//
#include <hip/hip_runtime.h>

// ---------------------------------------------------------------------------
// logp = -0.5*((x[t] - decay*x[t-1] - sigmoid(gain*s[t]))/noise)^2
//        - (log(noise) + 0.5*log(2*pi))
// B=4096, T=8192, f32. Pure streaming elementwise -> HBM-bound (~384MB moved,
// ~17us at 23.3 TB/s). One thread = 4 contiguous elements (b128 traffic).
// Wave32 ds_bpermute supplies x[i-1] across lanes; lane0 falls back to a
// scalar global load; row starts use 0.
// ---------------------------------------------------------------------------

typedef float v4f __attribute__((ext_vector_type(4)));

__device__ __forceinline__ float sigmoid_fast(float z) {
    // 1/(1+exp(-z)) : v_exp_f32 computes 2^x, so scale by log2(e); v_rcp_f32.
    float e = __builtin_amdgcn_exp2f(-z * 1.44269504088896340736f);
    return __builtin_amdgcn_rcpf(1.0f + e);
}

__global__ __launch_bounds__(256) void logistic_logp_kernel(
    const float* __restrict__ s,
    const float* __restrict__ x,
    const float* __restrict__ gain_p,
    const float* __restrict__ decay_p,
    const float* __restrict__ noise_p,
    float* __restrict__ out,
    long long n4,        // total float4s = B*T/4
    int t_mask)          // T-1 (T is a power of two)
{
    long long gid = (long long)blockIdx.x * blockDim.x + threadIdx.x;
    if (gid >= n4) return;
    long long base = gid << 2;            // first element index of this thread

    // --- uniform scalar parameters (single-element device arrays) ----------
    float gain  = gain_p[0];
    float decay = decay_p[0];
    float noise = noise_p[0];
    float inv_n = __builtin_amdgcn_rcpf(noise);
    float qc    = -0.5f * inv_n * inv_n;                        // -1/(2*noise^2)
    // log(noise) + 0.5*log(2*pi);  v_log_f32 is log2 -> scale by ln(2)
    float lognorm = __builtin_amdgcn_logf(noise) * 0.69314718055994530942f
                  + 0.91893853320467274178f;

    // --- prefetch both input streams ~64KB ahead (one req per 64B) ---------
    if ((threadIdx.x & 3) == 0) {
        long long pf = base + 16384;
        if (pf < (n4 << 2)) {
            __builtin_prefetch(s + pf, 0, 0);   // -> global_prefetch_b8
            __builtin_prefetch(x + pf, 0, 0);
        }
    }

    // --- streaming b128 loads, non-temporal (working set >> 192MB L2) ------
    v4f s4 = __builtin_nontemporal_load((const v4f*)(s + base));
    v4f x4 = __builtin_nontemporal_load((const v4f*)(x + base));

    int t0   = (int)(base & (long long)t_mask);   // column of first element
    int lane = (int)(threadIdx.x & 31u);          // wave32 on gfx1250

    // x[base-1]: lane L takes lane L-1's x4.w via wave32 backward permute.
    // (lane 0's index wraps harmlessly; it is overwritten below.)
    float xm1 = __int_as_float(
        __builtin_amdgcn_ds_bpermute((lane - 1) << 2, __float_as_int(x4.w)));
    if (lane == 0 && t0 != 0) xm1 = x[base - 1];  // cross-wave boundary
    if (t0 == 0) xm1 = 0.0f;                      // row start: x_prev = 0

    // --- compute 4 elements -------------------------------------------------
    float xp0 = xm1, xp1 = x4.x, xp2 = x4.y, xp3 = x4.z;

    float r0 = fmaf(-decay, xp0, x4.x) - sigmoid_fast(s4.x * gain);
    float r1 = fmaf(-decay, xp1, x4.y) - sigmoid_fast(s4.y * gain);
    float r2 = fmaf(-decay, xp2, x4.z) - sigmoid_fast(s4.z * gain);
    float r3 = fmaf(-decay, xp3, x4.w) - sigmoid_fast(s4.w * gain);

    v4f o;
    o.x = fmaf(r0 * r0, qc, -lognorm);
    o.y = fmaf(r1 * r1, qc, -lognorm);
    o.z = fmaf(r2 * r2, qc, -lognorm);
    o.w = fmaf(r3 * r3, qc, -lognorm);

    // --- write-once store, non-temporal ------------------------------------
    __builtin_nontemporal_store(o, (v4f*)(out + base));
}

extern "C" void kernel_launch(void* const* d_in, const int* in_sizes, int n_in,
                              void* d_out, int out_size, void* d_ws, size_t ws_size,
                              hipStream_t stream) {
    (void)n_in; (void)d_ws; (void)ws_size;
    // setup_inputs() order: s, x, gain, decay, noise
    const float* s     = (const float*)d_in[0];
    const float* x     = (const float*)d_in[1];
    const float* gain  = (const float*)d_in[2];
    const float* decay = (const float*)d_in[3];
    const float* noise = (const float*)d_in[4];
    float* out = (float*)d_out;

    const int T = 8192;                       // reference constant, power of 2
    long long n4 = (long long)out_size >> 2;  // 8,388,608 float4s
    const int threads = 256;                  // 8 wave32s per block
    unsigned blocks = (unsigned)((n4 + threads - 1) / threads);

    logistic_logp_kernel<<<blocks, threads, 0, stream>>>(
        s, x, gain, decay, noise, out, n4, T - 1);
}